// IPA_47442208751678
// MI455X (gfx1250) — compile-verified
//
#include <hip/hip_runtime.h>
#include <math.h>

// ---------------------------------------------------------------------------
// Invariant Point Attention for MI455X (gfx1250), fp32 WMMA pipeline.
// Roofline: pairwise (134MB) must stream twice -> ~11.5us @ 23.3 TB/s; total
// matrix FLOPs ~3.2G finish well under that even on the fp32 16x16x4 WMMA
// path, so we keep full fp32 precision and stay memory-bound.
// All inner loops are branch-free (no EXEC changes around v_wmma).
// ---------------------------------------------------------------------------

typedef __attribute__((ext_vector_type(2))) float v2f;
typedef __attribute__((ext_vector_type(8))) float v8f;

#define LQ    512
#define NH    12
#define NODE  384
#define PAIR  128
#define OUTIN 176   // 16 scaler + 128 pair + 24 point + 8 norm
#define KOUT  (NH*OUTIN)   // 2112

// workspace offsets (in floats)
#define OFF_SQ   ((size_t)0)
#define OFF_SK   (OFF_SQ  + (size_t)NH*LQ*16)
#define OFF_SV   (OFF_SK  + (size_t)NH*LQ*16)
#define OFF_PQ   (OFF_SV  + (size_t)NH*LQ*16)   // padded to 16 (12 real)
#define OFF_PK   (OFF_PQ  + (size_t)NH*LQ*16)
#define OFF_PV   (OFF_PK  + (size_t)NH*LQ*16)   // 24 per (h,l)
#define OFF_QN2  (OFF_PV  + (size_t)NH*LQ*24)
#define OFF_KN2  (OFF_QN2 + (size_t)NH*LQ)
#define OFF_BIAS (OFF_KN2 + (size_t)NH*LQ)      // [H][L][L]
#define OFF_ATTN (OFF_BIAS+ (size_t)NH*LQ*LQ)   // [H][L][L]
#define OFF_RES  (OFF_ATTN+ (size_t)NH*LQ*LQ)   // [L][H*176]  (K-contiguous)

// res element (head h, row l, feature f) in [L][H*176] layout
#define RES_IDX(h, l, f) (((size_t)(l)*NH + (h))*OUTIN + (f))

static __device__ __forceinline__ v8f wmma4(v2f a, v2f b, v8f c) {
  // D = A(16x4 f32) * B(4x16 f32) + C(16x16 f32)
  return __builtin_amdgcn_wmma_f32_16x16x4_f32(
      /*neg_a=*/false, a, /*neg_b=*/false, b,
      /*c_mod=*/(short)0, c, /*reuse_a=*/false, /*reuse_b=*/false);
}

static __device__ __forceinline__ v8f vzero8() {
  v8f z = {0.f,0.f,0.f,0.f,0.f,0.f,0.f,0.f};
  return z;
}

// ---------------------------------------------------------------------------
// K1: projections. grid = (32 M-tiles, 7 segments, 12 heads), 1 wave/block.
// seg: 0=sq(*0.25) 1=sk 2=sv 3=pq(pad16) 4=pk(pad16) 5=pv[0:16] 6=pv[16:24]
// Segment addressing hoisted out of the K-loop: base + stride + clamped col
// + 0/1 scale, so the inner loop is pure load -> wmma.
// ---------------------------------------------------------------------------
__global__ __launch_bounds__(32) void proj_kernel(
    const float* __restrict__ node,
    const float* __restrict__ sq_k, const float* __restrict__ sk_k,
    const float* __restrict__ sv_k, const float* __restrict__ pq_k,
    const float* __restrict__ pk_k, const float* __restrict__ pv_k,
    float* __restrict__ ws)
{
  const int lane  = threadIdx.x;
  const int mt    = blockIdx.x;
  const int seg   = blockIdx.y;
  const int h     = blockIdx.z;
  const int row   = lane & 15;
  const int khalf = (lane >> 4) * 2;

  // ---- per-segment B addressing (computed once) ----
  const float* wb = sk_k + (size_t)h*NODE*16;
  int   wstride = 16;
  int   wcol    = row;
  float wscale  = 1.f;
  switch (seg) {
    case 0: wb = sq_k + (size_t)h*NODE*16; wscale = 0.25f; break;
    case 1: break;
    case 2: wb = sv_k + (size_t)h*NODE*16; break;
    case 3: wb = pq_k + (size_t)h*NODE*12; wstride = 12;
            wcol = (row < 12) ? row : 11; wscale = (row < 12) ? 1.f : 0.f; break;
    case 4: wb = pk_k + (size_t)h*NODE*12; wstride = 12;
            wcol = (row < 12) ? row : 11; wscale = (row < 12) ? 1.f : 0.f; break;
    case 5: wb = pv_k + (size_t)h*NODE*24; wstride = 24; break;
    case 6: wb = pv_k + (size_t)h*NODE*24; wstride = 24;
            wcol = (row < 8) ? 16 + row : 23; wscale = (row < 8) ? 1.f : 0.f; break;
  }
  const float* arow = node + (size_t)(mt*16 + row)*NODE;
  const float* wp   = wb + wcol;

  v8f acc = vzero8();
  for (int k0 = 0; k0 < NODE; k0 += 4) {
    const int ka = k0 + khalf;
    v2f a; a.x = arow[ka]; a.y = arow[ka + 1];
    v2f b;
    b.x = wp[(size_t)ka * wstride]       * wscale;
    b.y = wp[(size_t)(ka + 1) * wstride] * wscale;
    acc = wmma4(a, b, acc);
  }

  // ---- store (D: VGPR r -> row r + 8*(lane>=16), col = lane&15) ----
  const int col   = lane & 15;
  const int rbase = (lane >> 4) * 8;
  float* ob = ws + OFF_SK + (size_t)h*LQ*16;
  int ostride = 16, ocol = col;
  bool dostore = true;
  switch (seg) {
    case 0: ob = ws + OFF_SQ + (size_t)h*LQ*16; break;
    case 1: break;
    case 2: ob = ws + OFF_SV + (size_t)h*LQ*16; break;
    case 3: ob = ws + OFF_PQ + (size_t)h*LQ*16; break;   // cols 12..15 store 0
    case 4: ob = ws + OFF_PK + (size_t)h*LQ*16; break;
    case 5: ob = ws + OFF_PV + (size_t)h*LQ*24; ostride = 24; break;
    case 6: ob = ws + OFF_PV + (size_t)h*LQ*24; ostride = 24;
            ocol = 16 + col; dostore = (col < 8); break;
  }
  if (dostore) {
#pragma unroll
    for (int r = 0; r < 8; ++r)
      ob[(size_t)(mt*16 + rbase + r)*ostride + ocol] = acc[r];
  }
}

// ---------------------------------------------------------------------------
// K2: rigid transform (global frame) + squared norms for the point distance.
// ---------------------------------------------------------------------------
__global__ void rigid_kernel(const float* __restrict__ R,
                             const float* __restrict__ t,
                             float* __restrict__ ws)
{
  const int idx = blockIdx.x * blockDim.x + threadIdx.x;
  if (idx >= NH * LQ) return;
  const int h = idx / LQ, l = idx % LQ;
  float Rm[9], tv[3];
#pragma unroll
  for (int i = 0; i < 9; ++i) Rm[i] = R[l*9 + i];
#pragma unroll
  for (int i = 0; i < 3; ++i) tv[i] = t[l*3 + i];

  float* Pq = ws + OFF_PQ + ((size_t)h*LQ + l)*16;
  float* Pk = ws + OFF_PK + ((size_t)h*LQ + l)*16;
  float* Pv = ws + OFF_PV + ((size_t)h*LQ + l)*24;

  float qs = 0.f, ksum = 0.f;
#pragma unroll
  for (int p = 0; p < 4; ++p) {
    float loc[3], g[3];
    for (int f = 0; f < 3; ++f) loc[f] = Pq[p*3+f];
    for (int gg = 0; gg < 3; ++gg) {
      g[gg] = tv[gg];
      for (int f = 0; f < 3; ++f) g[gg] += loc[f]*Rm[gg*3+f];
      Pq[p*3+gg] = g[gg]; qs += g[gg]*g[gg];
    }
    for (int f = 0; f < 3; ++f) loc[f] = Pk[p*3+f];
    for (int gg = 0; gg < 3; ++gg) {
      g[gg] = tv[gg];
      for (int f = 0; f < 3; ++f) g[gg] += loc[f]*Rm[gg*3+f];
      Pk[p*3+gg] = g[gg]; ksum += g[gg]*g[gg];
    }
  }
#pragma unroll
  for (int p = 0; p < 8; ++p) {
    float loc[3], g[3];
    for (int f = 0; f < 3; ++f) loc[f] = Pv[p*3+f];
    for (int gg = 0; gg < 3; ++gg) {
      g[gg] = tv[gg];
      for (int f = 0; f < 3; ++f) g[gg] += loc[f]*Rm[gg*3+f];
      Pv[p*3+gg] = g[gg];
    }
  }
  ws[OFF_QN2 + h*LQ + l] = qs;
  ws[OFF_KN2 + h*LQ + l] = ksum;
}

// ---------------------------------------------------------------------------
// K3: pair bias = (L*L x 128) @ (128 x H) as WMMA GEMM. Streams 134MB once.
// ---------------------------------------------------------------------------
__global__ __launch_bounds__(32) void bias_kernel(
    const float* __restrict__ pairwise, const float* __restrict__ pr_k,
    float* __restrict__ ws)
{
  float* bias = ws + OFF_BIAS;
  const int lane  = threadIdx.x;
  const int tile  = blockIdx.x;            // (L*L)/16 tiles
  const int row   = lane & 15;
  const int khalf = (lane >> 4) * 2;
  const int flat  = tile*16 + row;
  const float* prow = pairwise + (size_t)flat * PAIR;
  const float* prk  = pr_k + (size_t)((row < NH) ? row : NH - 1) * PAIR;
  const float bmask = (row < NH) ? 1.f : 0.f;

  v8f acc = vzero8();
  for (int k0 = 0; k0 < PAIR; k0 += 4) {
    const int ka = k0 + khalf;
    __builtin_prefetch(prow + ka + 64, 0, 0);
    v2f a; a.x = prow[ka];          a.y = prow[ka + 1];
    v2f b; b.x = prk[ka] * bmask;   b.y = prk[ka + 1] * bmask;
    acc = wmma4(a, b, acc);
  }
  const int col = lane & 15;               // head
  const int rbase = (lane >> 4) * 8;
  if (col < NH) {
#pragma unroll
    for (int r = 0; r < 8; ++r) {
      const int f2 = tile*16 + rbase + r;
      const int i = f2 >> 9, j = f2 & 511;
      bias[((size_t)col*LQ + i)*LQ + j] = acc[r];
    }
  }
}

// ---------------------------------------------------------------------------
// K4: attention core. block = (h, 16-row n-tile), 4 waves.
// Phase1 logits->LDS (WMMA), phase2 shuffle softmax, phase3 AV (WMMA).
// ---------------------------------------------------------------------------
#define SMS 516  // LDS row stride (floats), avoids 16-way bank conflicts
__global__ __launch_bounds__(128) void attn_kernel(
    const float* __restrict__ point_weight, float* __restrict__ ws)
{
  __shared__ float sm[16 * SMS];
  const int tid  = threadIdx.x;
  const int lane = tid & 31;
  const int w    = tid >> 5;
  const int nt   = blockIdx.x;
  const int h    = blockIdx.y;

  const float* Sq  = ws + OFF_SQ;  const float* Sk  = ws + OFF_SK;
  const float* Pq  = ws + OFF_PQ;  const float* Pk  = ws + OFF_PK;
  const float* qn2 = ws + OFF_QN2; const float* kn2 = ws + OFF_KN2;
  const float* bias= ws + OFF_BIAS;
  const float* Sv  = ws + OFF_SV;  const float* Pv  = ws + OFF_PV;
  float* attn = ws + OFF_ATTN;
  float* res  = ws + OFF_RES;

  const float pwx = point_weight[h];
  const float pw  = (pwx > 20.f) ? pwx : log1pf(__expf(pwx));     // softplus
  const float cpt = -0.5f * pw / sqrtf(18.f);   // point_depth_scale = sqrt(4*9/2)
  const float rs3 = 0.57735026919f;             // 1/sqrt(num_attn_logits=3)

  const int row   = lane & 15;
  const int khalf = (lane >> 4) * 2;
  const int col   = lane & 15;
  const int rbase = (lane >> 4) * 8;

  // preload A tiles (q rows of this n-tile): K=16 -> 4 wmma steps each
  v2f aq[4], ap[4];
#pragma unroll
  for (int ks = 0; ks < 4; ++ks) {
    const int k = ks*4 + khalf;
    const int l = nt*16 + row;
    aq[ks].x = Sq[((size_t)h*LQ + l)*16 + k];
    aq[ks].y = Sq[((size_t)h*LQ + l)*16 + k + 1];
    ap[ks].x = Pq[((size_t)h*LQ + l)*16 + k];
    ap[ks].y = Pq[((size_t)h*LQ + l)*16 + k + 1];
  }
  // |q|^2 for the 8 rows this lane's D registers map to (constant over mt)
  float qv[8];
#pragma unroll
  for (int r = 0; r < 8; ++r) qv[r] = qn2[h*LQ + nt*16 + rbase + r];

  for (int mt = w; mt < 32; mt += 4) {           // 8 m-tiles per wave
    v8f s = vzero8(), p = vzero8();
#pragma unroll
    for (int ks = 0; ks < 4; ++ks) {
      const int k = ks*4 + khalf;
      const int m = mt*16 + row;                 // B column indexes m
      v2f bq, bp;
      bq.x = Sk[((size_t)h*LQ + m)*16 + k]; bq.y = Sk[((size_t)h*LQ + m)*16 + k + 1];
      bp.x = Pk[((size_t)h*LQ + m)*16 + k]; bp.y = Pk[((size_t)h*LQ + m)*16 + k + 1];
      s = wmma4(aq[ks], bq, s);
      p = wmma4(ap[ks], bp, p);
    }
    const int m = mt*16 + col;                   // col fixed per lane
    const float kv = kn2[h*LQ + m];
#pragma unroll
    for (int r = 0; r < 8; ++r) {
      const int rr = rbase + r;
      const int n  = nt*16 + rr;
      const float dist = qv[r] + kv - 2.f*p[r];
      sm[rr*SMS + m] =
          (s[r] + bias[((size_t)h*LQ + n)*LQ + m] + cpt*dist) * rs3;
    }
  }
  __syncthreads();

  { // softmax: 8 lanes per row
    const int rrow = tid >> 3;
    const int sub  = tid & 7;
    float mx = -1e30f;
    for (int j = sub; j < LQ; j += 8) mx = fmaxf(mx, sm[rrow*SMS + j]);
    for (int o = 1; o < 8; o <<= 1) mx = fmaxf(mx, __shfl_xor(mx, o, 8));
    float sum = 0.f;
    for (int j = sub; j < LQ; j += 8) {
      const float e = __expf(sm[rrow*SMS + j] - mx);
      sm[rrow*SMS + j] = e; sum += e;
    }
    for (int o = 1; o < 8; o <<= 1) sum += __shfl_xor(sum, o, 8);
    const float inv = 1.f / sum;
    const int n = nt*16 + rrow;
    for (int j = sub; j < LQ; j += 8) {
      const float v = sm[rrow*SMS + j] * inv;
      sm[rrow*SMS + j] = v;
      attn[((size_t)h*LQ + n)*LQ + j] = v;
    }
  }
  __syncthreads();

  // wave0: attn@sv (16 cols); wave1: attn@pv[0:16]; wave2: attn@pv[16:24]
  if (w < 3) {
    // hoist wave-role B addressing out of the 128-step K loop
    const float* vb = (w == 0) ? Sv + (size_t)h*LQ*16 : Pv + (size_t)h*LQ*24;
    const int    vs = (w == 0) ? 16 : 24;
    int   vc = col;
    float vmask = 1.f;
    if (w == 2) { vc = (col < 8) ? 16 + col : 23; vmask = (col < 8) ? 1.f : 0.f; }
    const float* vp = vb + vc;

    v8f acc = vzero8();
    for (int kt = 0; kt < 128; ++kt) {
      const int k = kt*4 + khalf;
      v2f a; a.x = sm[row*SMS + k]; a.y = sm[row*SMS + k + 1];
      v2f b;
      b.x = vp[(size_t)k * vs]       * vmask;
      b.y = vp[(size_t)(k + 1) * vs] * vmask;
      acc = wmma4(a, b, acc);
    }
    const int fbase = (w == 0) ? 0 : 144;        // w==2 handled below
#pragma unroll
    for (int r = 0; r < 8; ++r) {
      const int n = nt*16 + rbase + r;
      const float v = acc[r];
      if (w < 2)        res[RES_IDX(h, n, fbase + col)] = v;
      else if (col < 8) res[RES_IDX(h, n, 160 + col)]   = v;
    }
  }
}

// ---------------------------------------------------------------------------
// K5: pairwise output: per row n, (12x512 attn) @ (512x128 pairwise[n]).
// One wave handles all 8 N-tiles so attn is loaded once. Streams 134MB (L2).
// ---------------------------------------------------------------------------
__global__ __launch_bounds__(32) void pairout_kernel(
    const float* __restrict__ pairwise, float* __restrict__ ws)
{
  const float* attn = ws + OFF_ATTN;
  float* res = ws + OFF_RES;
  const int lane  = threadIdx.x;
  const int n     = blockIdx.x;
  const int row   = lane & 15;    // head (A row)
  const int khalf = (lane >> 4) * 2;
  const int col   = lane & 15;    // B column within tile

  const int hh0 = (row < NH) ? row : NH - 1;
  const float amask = (row < NH) ? 1.f : 0.f;
  const float* arow = attn + ((size_t)hh0*LQ + n)*LQ;

  v8f acc[8];
#pragma unroll
  for (int i = 0; i < 8; ++i) acc[i] = vzero8();

  for (int kt = 0; kt < 128; ++kt) {
    const int m = kt*4 + khalf;
    v2f a;
    a.x = arow[m] * amask;
    a.y = arow[m + 1] * amask;
    const float* pr0 = pairwise + ((size_t)n*LQ + m) * PAIR + col;
    __builtin_prefetch(pr0 + 4*PAIR, 0, 0);
#pragma unroll
    for (int ntile = 0; ntile < 8; ++ntile) {
      v2f b; b.x = pr0[ntile*16]; b.y = pr0[ntile*16 + PAIR];
      acc[ntile] = wmma4(a, b, acc[ntile]);
    }
  }
  const int rbase = (lane >> 4) * 8;
#pragma unroll
  for (int ntile = 0; ntile < 8; ++ntile)
#pragma unroll
    for (int r = 0; r < 8; ++r) {
      const int hh = rbase + r;
      if (hh < NH)
        res[RES_IDX(hh, n, 16 + ntile*16 + col)] = acc[ntile][r];
    }
}

// ---------------------------------------------------------------------------
// K6: point output back to local frame (R^T applied) + point norms.
// ---------------------------------------------------------------------------
__global__ void pfinish_kernel(const float* __restrict__ R,
                               const float* __restrict__ t,
                               float* __restrict__ ws)
{
  const int idx = blockIdx.x * blockDim.x + threadIdx.x;
  if (idx >= NH * LQ) return;
  const int h = idx / LQ, l = idx % LQ;
  float* resp = ws + OFF_RES + RES_IDX(h, l, 0);
  float Rm[9], tv[3];
#pragma unroll
  for (int i = 0; i < 9; ++i) Rm[i] = R[l*9 + i];
#pragma unroll
  for (int i = 0; i < 3; ++i) tv[i] = t[l*3 + i];
#pragma unroll
  for (int p = 0; p < 8; ++p) {
    float g[3], loc[3], nrm = 0.f;
    for (int gg = 0; gg < 3; ++gg) g[gg] = resp[144 + p*3 + gg] - tv[gg];
    for (int f = 0; f < 3; ++f) {
      float v = 0.f;
      for (int gg = 0; gg < 3; ++gg) v += g[gg]*Rm[gg*3 + f];
      loc[f] = v; nrm += v*v;
    }
    for (int f = 0; f < 3; ++f) resp[144 + p*3 + f] = loc[f];
    resp[168 + p] = sqrtf(nrm + 1e-12f);
  }
}

// ---------------------------------------------------------------------------
// K7: output projection: (512 x 2112) @ (2112 x 384) + bias.
// res is [L][2112] so A is K-contiguous; out_k (H,176,384) flattens to
// (2112,384) row-major so B index is just kk*384+o. No div/mod in the loop.
// ---------------------------------------------------------------------------
__global__ __launch_bounds__(32) void outproj_kernel(
    const float* __restrict__ out_k, const float* __restrict__ out_b,
    const float* __restrict__ res, float* __restrict__ out)
{
  const int lane  = threadIdx.x;
  const int mt    = blockIdx.x;    // 32 l-tiles
  const int ntile = blockIdx.y;    // 24 o-tiles
  const int row   = lane & 15;
  const int khalf = (lane >> 4) * 2;
  const int col   = lane & 15;
  const int o     = ntile*16 + col;

  const float* resl = res + (size_t)(mt*16 + row) * KOUT;
  const float* okp  = out_k + o;

  v8f acc = vzero8();
  for (int k0 = 0; k0 < KOUT; k0 += 4) {
    const int ka = k0 + khalf;
    v2f a; a.x = resl[ka];                 a.y = resl[ka + 1];
    v2f b; b.x = okp[(size_t)ka * NODE];   b.y = okp[(size_t)(ka + 1) * NODE];
    acc = wmma4(a, b, acc);
  }
  const int rbase = (lane >> 4) * 8;
  const float ob = out_b[o];
#pragma unroll
  for (int r = 0; r < 8; ++r) {
    const int l = mt*16 + rbase + r;
    out[(size_t)l*NODE + o] = acc[r] + ob;
  }
}

// ---------------------------------------------------------------------------
extern "C" void kernel_launch(void* const* d_in, const int* in_sizes, int n_in,
                              void* d_out, int out_size, void* d_ws, size_t ws_size,
                              hipStream_t stream) {
  const float* node  = (const float*)d_in[0];
  const float* R     = (const float*)d_in[1];
  const float* t     = (const float*)d_in[2];
  const float* pairw = (const float*)d_in[3];
  const float* sq_k  = (const float*)d_in[4];
  const float* sk_k  = (const float*)d_in[5];
  const float* sv_k  = (const float*)d_in[6];
  const float* pq_k  = (const float*)d_in[7];
  const float* pk_k  = (const float*)d_in[8];
  const float* pv_k  = (const float*)d_in[9];
  const float* pr_k  = (const float*)d_in[10];
  const float* out_k = (const float*)d_in[11];
  const float* out_b = (const float*)d_in[12];
  const float* pwgt  = (const float*)d_in[13];
  float* ws  = (float*)d_ws;     // needs ~32.1 MB
  float* out = (float*)d_out;

  proj_kernel<<<dim3(32, 7, NH), 32, 0, stream>>>(
      node, sq_k, sk_k, sv_k, pq_k, pk_k, pv_k, ws);
  rigid_kernel<<<(NH*LQ + 127)/128, 128, 0, stream>>>(R, t, ws);
  bias_kernel<<<(LQ*LQ)/16, 32, 0, stream>>>(pairw, pr_k, ws);
  attn_kernel<<<dim3(32, NH), 128, 0, stream>>>(pwgt, ws);
  pairout_kernel<<<LQ, 32, 0, stream>>>(pairw, ws);
  pfinish_kernel<<<(NH*LQ + 127)/128, 128, 0, stream>>>(R, t, ws);
  outproj_kernel<<<dim3(32, 24), 32, 0, stream>>>(out_k, out_b, ws + OFF_RES, out);
}